// GraphormerBlock_70789650972751
// MI455X (gfx1250) — compile-verified
//
#include <hip/hip_runtime.h>
#include <math.h>

// ---------------------------------------------------------------------------
// Graphormer block for MI455X (gfx1250, wave32).
//  - All GEMMs on V_WMMA_F32_16X16X4_F32 (fp32-exact vs the fp32 reference;
//    the workload is HBM/scatter bound so fp32 matrix rate is sufficient).
//  - A-tiles staged into LDS via GLOBAL_LOAD_ASYNC_TO_LDS_B128 (ASYNCcnt).
//  - Segment softmax via L2-resident fp32 atomics (amax/denom/x_attn << 192MB L2).
// ---------------------------------------------------------------------------

typedef float v2f __attribute__((ext_vector_type(2)));
typedef float v8f __attribute__((ext_vector_type(8)));

__device__ __forceinline__ float warp_sum(float v) {
  v += __shfl_xor(v, 16, 32);
  v += __shfl_xor(v, 8, 32);
  v += __shfl_xor(v, 4, 32);
  v += __shfl_xor(v, 2, 32);
  v += __shfl_xor(v, 1, 32);
  return v;
}

__device__ __forceinline__ void atomicMaxF(float* addr, float val) {
  // int-pattern trick -> global_atomic_max_i32 / global_atomic_min_u32;
  // correct for mixed signs with -inf init.
  if (val >= 0.0f) atomicMax((int*)addr, __float_as_int(val));
  else             atomicMin((unsigned int*)addr, __float_as_uint(val));
}

// async global->LDS 16B move: VDST = wave-relative LDS byte address (low 32
// bits of the generic pointer, per the flat->LDS aperture mapping), VADDR =
// 64-bit global address. Tracked by ASYNCcnt.
__device__ __forceinline__ void async_g2l_b128(void* lds_ptr, const void* gptr) {
  unsigned lds = (unsigned)(uintptr_t)lds_ptr;
  unsigned long long ga = (unsigned long long)(uintptr_t)gptr;
  asm volatile("global_load_async_to_lds_b128 %0, %1, off"
               :: "v"(lds), "v"(ga) : "memory");
}
__device__ __forceinline__ void async_wait0() {
  asm volatile("s_wait_asynccnt 0x0" ::: "memory");
}

// ---------------------------------------------------------------------------
// init: deg = 0, amax = -inf, denom = 0
__global__ void k_init(int* deg, float* amax, float* denom, int n, int nh) {
  int i = blockIdx.x * blockDim.x + threadIdx.x;
  if (i < n) deg[i] = 0;
  if (i < nh) { amax[i] = -INFINITY; denom[i] = 0.0f; }
}

// out-degree via scatter-add on src
__global__ void k_degree(const int* __restrict__ src, int* __restrict__ deg, int E) {
  int i = blockIdx.x * blockDim.x + threadIdx.x;
  if (i < E) atomicAdd(&deg[src[i]], 1);
}

// x_in = x + deg_emb[clip(deg,0,511)]; h = LN(x_in).  One wave per node row.
__global__ void k_xin_ln(const float* __restrict__ x, const int* __restrict__ deg,
                         const float* __restrict__ demb,
                         const float* __restrict__ g, const float* __restrict__ b,
                         float* __restrict__ x_in, float* __restrict__ h, int n) {
  int w = (blockIdx.x * blockDim.x + threadIdx.x) >> 5;
  int lane = threadIdx.x & 31;
  if (w >= n) return;
  int dg = deg[w]; dg = dg < 0 ? 0 : (dg > 511 ? 511 : dg);
  float4 xv = *(const float4*)(x + (size_t)w * 128 + lane * 4);
  float4 dv = *(const float4*)(demb + (size_t)dg * 128 + lane * 4);
  float4 t;
  t.x = xv.x + dv.x; t.y = xv.y + dv.y; t.z = xv.z + dv.z; t.w = xv.w + dv.w;
  *(float4*)(x_in + (size_t)w * 128 + lane * 4) = t;
  float s  = t.x + t.y + t.z + t.w;
  float s2 = t.x * t.x + t.y * t.y + t.z * t.z + t.w * t.w;
  s = warp_sum(s); s2 = warp_sum(s2);
  float mean = s * (1.0f / 128.0f);
  float var  = s2 * (1.0f / 128.0f) - mean * mean;
  float rstd = rsqrtf(var + 1e-5f);
  float4 gv = *(const float4*)(g + lane * 4);
  float4 bv = *(const float4*)(b + lane * 4);
  float4 o;
  o.x = (t.x - mean) * rstd * gv.x + bv.x;
  o.y = (t.y - mean) * rstd * gv.y + bv.y;
  o.z = (t.z - mean) * rstd * gv.z + bv.z;
  o.w = (t.w - mean) * rstd * gv.w + bv.w;
  *(float4*)(h + (size_t)w * 128 + lane * 4) = o;
}

// generic 128-wide LayerNorm, one wave per row
__global__ void k_ln128(const float* __restrict__ in,
                        const float* __restrict__ g, const float* __restrict__ b,
                        float* __restrict__ out, int n) {
  int w = (blockIdx.x * blockDim.x + threadIdx.x) >> 5;
  int lane = threadIdx.x & 31;
  if (w >= n) return;
  float4 t = *(const float4*)(in + (size_t)w * 128 + lane * 4);
  float s  = t.x + t.y + t.z + t.w;
  float s2 = t.x * t.x + t.y * t.y + t.z * t.z + t.w * t.w;
  s = warp_sum(s); s2 = warp_sum(s2);
  float mean = s * (1.0f / 128.0f);
  float var  = s2 * (1.0f / 128.0f) - mean * mean;
  float rstd = rsqrtf(var + 1e-5f);
  float4 gv = *(const float4*)(g + lane * 4);
  float4 bv = *(const float4*)(b + lane * 4);
  float4 o;
  o.x = (t.x - mean) * rstd * gv.x + bv.x;
  o.y = (t.y - mean) * rstd * gv.y + bv.y;
  o.z = (t.z - mean) * rstd * gv.z + bv.z;
  o.w = (t.w - mean) * rstd * gv.w + bv.w;
  *(float4*)(out + (size_t)w * 128 + lane * 4) = o;
}

// ---------------------------------------------------------------------------
// Generic fp32 WMMA GEMM:  C[M,N] = act(A[M,K] @ B[K,N] + bias) + R
//   block = 256 threads (8 waves); 16-row M tile staged into LDS with
//   GLOBAL_LOAD_ASYNC_TO_LDS_B128 (row stride K+4 floats; rows are 16B
//   multiples so chunks stay aligned); each wave owns 16 output cols.
//   M, K multiples of 16 (true for all launches here).
// act: 0 = none, 1 = exact GELU, 2 = ReLU
__global__ __launch_bounds__(256)
void k_gemm_wmma(const float* __restrict__ A, const float* __restrict__ B,
                 const float* __restrict__ bias, const float* __restrict__ R,
                 float* __restrict__ C, int M, int N, int K, int act) {
  extern __shared__ float At[];            // 16 x (K+4)
  const int KP = K + 4;
  const int tid = threadIdx.x;
  const int row0 = blockIdx.x * 16;
  // async stage: 16 rows x (K/4) 16B chunks, uniform trip count -> EXEC all 1s
  const int chunksPerRow = K >> 2;
  for (int i = tid; i < 16 * chunksPerRow; i += 256) {
    int r = i / chunksPerRow, c4 = (i - r * chunksPerRow) << 2;
    async_g2l_b128(&At[r * KP + c4], A + (size_t)(row0 + r) * K + c4);
  }
  async_wait0();
  __syncthreads();
  const int wave = tid >> 5, lane = tid & 31;
  const int half = lane >> 4, mn = lane & 15;
  const int ntiles = N >> 4;
  for (int nt = wave; nt < ntiles; nt += 8) {
    const int col = nt * 16 + mn;
    v8f acc = {0.f, 0.f, 0.f, 0.f, 0.f, 0.f, 0.f, 0.f};
    for (int kk = 0; kk < K; kk += 4) {
      // A frag: lanes 0-15 -> K = kk,kk+1 ; lanes 16-31 -> K = kk+2,kk+3
      // even 8B-aligned index -> single ds_load_b64
      v2f a = *(const v2f*)&At[mn * KP + kk + 2 * half];
      v2f bb;
      bb[0] = B[(size_t)(kk + 2 * half) * N + col];
      bb[1] = B[(size_t)(kk + 2 * half + 1) * N + col];
      acc = __builtin_amdgcn_wmma_f32_16x16x4_f32(false, a, false, bb,
                                                  (short)0, acc, false, false);
    }
    const float bia = bias ? bias[col] : 0.0f;
    for (int r = 0; r < 8; ++r) {
      const int m = row0 + r + 8 * half;   // C/D layout: VGPR r <-> row r+8*half
      float v = acc[r] + bia;
      if (act == 1)      v = 0.5f * v * (1.0f + erff(v * 0.7071067811865475f));
      else if (act == 2) v = v > 0.0f ? v : 0.0f;
      if (R) v += R[(size_t)m * N + col];
      C[(size_t)m * N + col] = v;
    }
  }
}

// ---------------------------------------------------------------------------
// attention scores: one wave per edge; lane covers 4 channels; 4 lanes = 1 head
__global__ void k_attn_score(const float* __restrict__ q, const float* __restrict__ k,
                             const float* __restrict__ e,
                             const int* __restrict__ src, const int* __restrict__ dst,
                             float* __restrict__ alpha, float* __restrict__ amax, int E) {
  int w = (blockIdx.x * blockDim.x + threadIdx.x) >> 5;
  int lane = threadIdx.x & 31;
  if (w >= E) return;
  int s = src[w], d = dst[w];
  float4 qv = *(const float4*)(q + (size_t)d * 128 + lane * 4);
  float4 kv = *(const float4*)(k + (size_t)s * 128 + lane * 4);
  float4 ev = *(const float4*)(e + (size_t)w * 128 + lane * 4);
  float p = qv.x * (kv.x + ev.x) + qv.y * (kv.y + ev.y) +
            qv.z * (kv.z + ev.z) + qv.w * (kv.w + ev.w);
  p += __shfl_xor(p, 1, 32);
  p += __shfl_xor(p, 2, 32);
  p *= 0.25f;                               // 1/sqrt(C), C = 16
  if ((lane & 3) == 0) {
    int hh = lane >> 2;
    alpha[(size_t)w * 8 + hh] = p;
    atomicMaxF(&amax[(size_t)d * 8 + hh], p);
  }
}

__global__ void k_amax_fix(float* amax, int n) {
  int i = blockIdx.x * blockDim.x + threadIdx.x;
  if (i < n) {
    float m = amax[i];
    if (!(m > -1e38f && m < 1e38f)) amax[i] = 0.0f;  // -inf / nan -> 0
  }
}

__global__ void k_attn_exp(float* __restrict__ alpha, const float* __restrict__ amax,
                           const int* __restrict__ dst, float* __restrict__ denom, int total) {
  int i = blockIdx.x * blockDim.x + threadIdx.x;
  if (i >= total) return;
  int e = i >> 3, h = i & 7;
  int d = dst[e];
  float ex = expf(alpha[i] - amax[(size_t)d * 8 + h]);
  alpha[i] = ex;
  atomicAdd(&denom[(size_t)d * 8 + h], ex);
}

__global__ void k_xattn_init(const float* __restrict__ xin, const float* __restrict__ hs,
                             float* __restrict__ xat, int n) {
  int i = blockIdx.x * blockDim.x + threadIdx.x;
  if (i < n) xat[i] = xin[i] + hs[i];
}

// message scatter: one wave per edge; 4 fp32 atomics per lane into L2-resident x_attn
__global__ void k_attn_scatter(const float* __restrict__ v, const float* __restrict__ e,
                               const float* __restrict__ ex, const float* __restrict__ denom,
                               const int* __restrict__ src, const int* __restrict__ dst,
                               float* __restrict__ xattn, int E) {
  int w = (blockIdx.x * blockDim.x + threadIdx.x) >> 5;
  int lane = threadIdx.x & 31;
  if (w >= E) return;
  int s = src[w], d = dst[w];
  int hh = lane >> 2;
  float wt = ex[(size_t)w * 8 + hh] / (denom[(size_t)d * 8 + hh] + 1e-16f);
  float4 vv = *(const float4*)(v + (size_t)s * 128 + lane * 4);
  float4 ev = *(const float4*)(e + (size_t)w * 128 + lane * 4);
  float* o = xattn + (size_t)d * 128 + lane * 4;
  atomicAdd(o + 0, (vv.x + ev.x) * wt);
  atomicAdd(o + 1, (vv.y + ev.y) * wt);
  atomicAdd(o + 2, (vv.z + ev.z) * wt);
  atomicAdd(o + 3, (vv.w + ev.w) * wt);
}

// ---------------------------------------------------------------------------
// edge-update GEMM: raw = relu([xn[src], xn[dst], ea] @ Wm + bm), K=320, N=128
// gathered A tile staged with async global->LDS b128 (all source rows are
// 16B-chunk aligned: xn rows 512B, ea rows 256B).
__global__ __launch_bounds__(256)
void k_edge_gemm(const float* __restrict__ xn, const float* __restrict__ ea,
                 const int* __restrict__ src, const int* __restrict__ dst,
                 const float* __restrict__ Wm, const float* __restrict__ bm,
                 float* __restrict__ raw, int E) {
  const int K = 320, KP = 324, N = 128;
  __shared__ float At[16 * 324];           // ~20.7 KB of the 320 KB WGP LDS
  __shared__ int sIdx[16], dIdx[16];
  const int tid = threadIdx.x;
  const int e0 = blockIdx.x * 16;
  if (tid < 16) { sIdx[tid] = src[e0 + tid]; dIdx[tid] = dst[e0 + tid]; }
  __syncthreads();
  // 16 rows x 80 chunks of 16B, 1280/256 = 5 iterations, uniform trip count
  for (int i = tid; i < 16 * 80; i += 256) {
    int r = i / 80, c4 = (i - r * 80) << 2;
    const float* g;
    if (c4 < 128)      g = xn + (size_t)sIdx[r] * 128 + c4;
    else if (c4 < 256) g = xn + (size_t)dIdx[r] * 128 + (c4 - 128);
    else               g = ea + (size_t)(e0 + r) * 64 + (c4 - 256);
    async_g2l_b128(&At[r * KP + c4], g);
  }
  async_wait0();
  __syncthreads();
  const int wave = tid >> 5, lane = tid & 31;
  const int half = lane >> 4, mn = lane & 15;
  const int col = wave * 16 + mn;          // 8 waves cover the 128 output cols
  v8f acc = {0.f, 0.f, 0.f, 0.f, 0.f, 0.f, 0.f, 0.f};
  for (int kk = 0; kk < K; kk += 4) {
    v2f a = *(const v2f*)&At[mn * KP + kk + 2 * half];
    v2f bb;
    bb[0] = Wm[(size_t)(kk + 2 * half) * N + col];
    bb[1] = Wm[(size_t)(kk + 2 * half + 1) * N + col];
    acc = __builtin_amdgcn_wmma_f32_16x16x4_f32(false, a, false, bb,
                                                (short)0, acc, false, false);
  }
  const float bia = bm[col];
  for (int r = 0; r < 8; ++r) {
    const int m = e0 + r + 8 * half;
    float v = acc[r] + bia;
    v = v > 0.0f ? v : 0.0f;               // relu
    raw[(size_t)m * N + col] = v;
  }
}

// ea_new = LN(ea + sigmoid(gate) * delta) over 64 dims; one wave per edge
__global__ void k_edge_update(const float* __restrict__ raw, const float* __restrict__ ea,
                              const float* __restrict__ g, const float* __restrict__ b,
                              float* __restrict__ out, int E) {
  int w = (blockIdx.x * blockDim.x + threadIdx.x) >> 5;
  int lane = threadIdx.x & 31;
  if (w >= E) return;
  float2 dv = *(const float2*)(raw + (size_t)w * 128 + lane * 2);        // delta
  float2 gt = *(const float2*)(raw + (size_t)w * 128 + 64 + lane * 2);   // gate
  float2 av = *(const float2*)(ea + (size_t)w * 64 + lane * 2);
  float t0 = av.x + dv.x / (1.0f + expf(-gt.x));
  float t1 = av.y + dv.y / (1.0f + expf(-gt.y));
  float s  = t0 + t1;
  float s2 = t0 * t0 + t1 * t1;
  s = warp_sum(s); s2 = warp_sum(s2);
  float mean = s * (1.0f / 64.0f);
  float var  = s2 * (1.0f / 64.0f) - mean * mean;
  float rstd = rsqrtf(var + 1e-5f);
  float2 gv = *(const float2*)(g + lane * 2);
  float2 bv = *(const float2*)(b + lane * 2);
  float2 o;
  o.x = (t0 - mean) * rstd * gv.x + bv.x;
  o.y = (t1 - mean) * rstd * gv.y + bv.y;
  *(float2*)(out + (size_t)w * 64 + lane * 2) = o;
}

// ---------------------------------------------------------------------------
extern "C" void kernel_launch(void* const* d_in, const int* in_sizes, int n_in,
                              void* d_out, int out_size, void* d_ws, size_t ws_size,
                              hipStream_t stream) {
  const float* x    = (const float*)d_in[0];
  const int*   ei   = (const int*)d_in[1];
  const float* ea   = (const float*)d_in[2];
  const float* demb = (const float*)d_in[3];
  const float* ln_g = (const float*)d_in[4];
  const float* ln_b = (const float*)d_in[5];
  const float* Wq = (const float*)d_in[6],  *bq = (const float*)d_in[7];
  const float* Wk = (const float*)d_in[8],  *bk = (const float*)d_in[9];
  const float* Wv = (const float*)d_in[10], *bv = (const float*)d_in[11];
  const float* We = (const float*)d_in[12], *be = (const float*)d_in[13];
  const float* Ws = (const float*)d_in[14], *bs = (const float*)d_in[15];
  const float* W1 = (const float*)d_in[16], *b1 = (const float*)d_in[17];
  const float* W2 = (const float*)d_in[18], *b2 = (const float*)d_in[19];
  const float* eug = (const float*)d_in[20], *eub = (const float*)d_in[21];
  const float* Wm = (const float*)d_in[22], *bm = (const float*)d_in[23];
  const float* elg = (const float*)d_in[24], *elb = (const float*)d_in[25];

  const int N = in_sizes[0] / 128;   // 50000
  const int E = in_sizes[1] / 2;     // 800000
  const int* src = ei;
  const int* dst = ei + E;

  // ---- workspace carve (all offsets 256B aligned) ----
  char* ws = (char*)d_ws;
  size_t off = 0;
  auto carve = [&](size_t bytes) -> void* {
    void* p = ws + off;
    off += (bytes + 255) & ~(size_t)255;
    return p;
  };
  int*   deg   = (int*)carve((size_t)N * 4);
  float* x_in  = (float*)carve((size_t)N * 128 * 4);
  float* hbuf  = (float*)carve((size_t)N * 128 * 4);      // h, then hf, then xn
  float* qb    = (float*)carve((size_t)N * 128 * 4 * 4);  // q|k|v|skip; later FFN mid [N,512]
  float* kb    = qb + (size_t)N * 128;
  float* vb    = kb + (size_t)N * 128;
  float* hs    = vb + (size_t)N * 128;
  float* mid   = qb;                                      // alias (q/k/v/skip dead by FFN)
  float* ebuf  = (float*)carve((size_t)E * 128 * 4);      // e; later raw
  float* wbuf  = (float*)carve((size_t)E * 8 * 4);        // alpha -> ex
  float* amax  = (float*)carve((size_t)N * 8 * 4);
  float* denom = (float*)carve((size_t)N * 8 * 4);
  float* xat   = (float*)carve((size_t)N * 128 * 4);
  (void)ws_size; (void)n_in; (void)out_size;

  float* xnew = (float*)d_out;              // [N,128]
  float* eout = xnew + (size_t)N * 128;     // [E,64]

  const dim3 blk(256);
  auto cdiv = [](long a, long b) { return (int)((a + b - 1) / b); };
  const size_t sh64  = 16 * (64 + 4) * sizeof(float);
  const size_t sh128 = 16 * (128 + 4) * sizeof(float);
  const size_t sh512 = 16 * (512 + 4) * sizeof(float);

  k_init<<<cdiv((long)N * 8, 256), blk, 0, stream>>>(deg, amax, denom, N, N * 8);
  k_degree<<<cdiv(E, 256), blk, 0, stream>>>(src, deg, E);
  k_xin_ln<<<cdiv((long)N * 32, 256), blk, 0, stream>>>(x, deg, demb, ln_g, ln_b, x_in, hbuf, N);

  k_gemm_wmma<<<N / 16, blk, sh128, stream>>>(hbuf, Wq, bq, nullptr, qb, N, 128, 128, 0);
  k_gemm_wmma<<<N / 16, blk, sh128, stream>>>(hbuf, Wk, bk, nullptr, kb, N, 128, 128, 0);
  k_gemm_wmma<<<N / 16, blk, sh128, stream>>>(hbuf, Wv, bv, nullptr, vb, N, 128, 128, 0);
  k_gemm_wmma<<<N / 16, blk, sh128, stream>>>(hbuf, Ws, bs, nullptr, hs, N, 128, 128, 0);
  k_gemm_wmma<<<E / 16, blk, sh64, stream>>>(ea, We, be, nullptr, ebuf, E, 128, 64, 0);

  k_attn_score<<<cdiv((long)E * 32, 256), blk, 0, stream>>>(qb, kb, ebuf, src, dst, wbuf, amax, E);
  k_amax_fix<<<cdiv((long)N * 8, 256), blk, 0, stream>>>(amax, N * 8);
  k_attn_exp<<<cdiv((long)E * 8, 256), blk, 0, stream>>>(wbuf, amax, dst, denom, E * 8);
  k_xattn_init<<<cdiv((long)N * 128, 256), blk, 0, stream>>>(x_in, hs, xat, N * 128);
  k_attn_scatter<<<cdiv((long)E * 32, 256), blk, 0, stream>>>(vb, ebuf, wbuf, denom, src, dst, xat, E);

  k_ln128<<<cdiv((long)N * 32, 256), blk, 0, stream>>>(xat, ln_g, ln_b, hbuf, N);
  k_gemm_wmma<<<N / 16, blk, sh128, stream>>>(hbuf, W1, b1, nullptr, mid, N, 512, 128, 1);
  k_gemm_wmma<<<N / 16, blk, sh512, stream>>>(mid, W2, b2, xat, xnew, N, 128, 512, 0);

  k_ln128<<<cdiv((long)N * 32, 256), blk, 0, stream>>>(xnew, eug, eub, hbuf, N);
  k_edge_gemm<<<E / 16, blk, 0, stream>>>(hbuf, ea, src, dst, Wm, bm, ebuf, E);
  k_edge_update<<<cdiv((long)E * 32, 256), blk, 0, stream>>>(ebuf, ea, elg, elb, eout, E);
}